// GraphSAGE_9139690406075
// MI455X (gfx1250) — compile-verified
//
#include <hip/hip_runtime.h>
#include <stdint.h>

#define N_NODES 50000
#define D_IN 128
#define D_H 128
#define D_OUTF 64

typedef float v2f __attribute__((ext_vector_type(2)));
typedef float v8f __attribute__((ext_vector_type(8)));

#if __has_builtin(__builtin_amdgcn_global_load_async_to_lds_b128) && \
    __has_builtin(__builtin_amdgcn_s_wait_asynccnt)
#define USE_ASYNC_LDS 1
typedef int v4i_ __attribute__((vector_size(16)));
typedef __attribute__((address_space(1))) v4i_ g_v4i;   // global (AS1)
typedef __attribute__((address_space(3))) v4i_ l_v4i;   // LDS (AS3)
#else
#define USE_ASYNC_LDS 0
#endif

// -------------------- utility: zero a float buffer --------------------
__global__ void zero_kernel(float* __restrict__ p, int n) {
    int i = blockIdx.x * blockDim.x + threadIdx.x;
    if (i < n) p[i] = 0.0f;
}

// -------------------- in-degree via float atomics --------------------
__global__ void degree_kernel(const int* __restrict__ dst, float* __restrict__ deg, int E) {
    int e = blockIdx.x * blockDim.x + threadIdx.x;
    if (e < E) atomicAdd(&deg[dst[e]], 1.0f);
}

// deg[i] <- 1 / max(deg[i], 1)
__global__ void recip_kernel(float* __restrict__ deg, int n) {
    int i = blockIdx.x * blockDim.x + threadIdx.x;
    if (i < n) deg[i] = 1.0f / fmaxf(deg[i], 1.0f);
}

// -------------------- WMMA f32 GEMM: Z = X(M x K) @ W(K x N) [+ bias] --------------------
// Block = (N/16) wave32s. tileM = blockIdx.x, tileN = waveId. The 16xK X tile shared by
// all waves of the block is staged once into LDS (async global->LDS on gfx1250), then the
// fully-unrolled K loop issues V_WMMA_F32_16X16X4_F32 with immediate-offset operand loads.
// A(16x4): lanes 0-15 hold M=lane, K={0,1}; lanes 16-31 hold M=lane-16, K={2,3}.
// B(4x16): lanes 0-15 hold N=lane rows K={0,1}; lanes 16-31 rows K={2,3}.
// C/D(16x16): VGPR r -> M = r + 8*(lane>=16), N = lane&15.
template <int K, int N>
__global__ void gemm_wmma_f32(const float* __restrict__ X,
                              const float* __restrict__ W,
                              const float* __restrict__ bias,
                              float* __restrict__ Z) {
    constexpr int WAVES  = N / 16;      // waves per block == tiles in N
    constexpr int NT     = WAVES * 32;  // threads per block
    constexpr int STRIDE = K + 4;       // LDS row stride (floats), bank-conflict padding

    __shared__ float sX[16 * STRIDE];

    const int tid    = threadIdx.x;
    const int waveId = tid >> 5;
    const int lane   = tid & 31;
    const int tileM  = blockIdx.x;
    const int tileN  = waveId;

    // ---- cooperative stage of the 16 x K X tile into LDS (float4 granularity) ----
    constexpr int vecPerRow = K / 4;
    constexpr int nVec      = 16 * vecPerRow;
    const float* __restrict__ Xtile = X + (size_t)tileM * 16 * K;
#pragma unroll
    for (int v = tid; v < nVec; v += NT) {
        const int r = v / vecPerRow;
        const int c = (v % vecPerRow) * 4;
#if USE_ASYNC_LDS
        __builtin_amdgcn_global_load_async_to_lds_b128(
            (g_v4i*)(uintptr_t)(Xtile + (size_t)r * K + c),
            (l_v4i*)(uint32_t)(uintptr_t)(&sX[r * STRIDE + c]), 0, 0);
#else
        *(float4*)(&sX[r * STRIDE + c]) = *(const float4*)(Xtile + (size_t)r * K + c);
#endif
    }
#if USE_ASYNC_LDS
    __builtin_amdgcn_s_wait_asynccnt(0);
#endif
    __syncthreads();

    // ---- 16x16 tile product, K unrolled: immediate-offset ds/global loads ----
    const int half = lane >> 4;   // 0: K pair {0,1}; 1: K pair {2,3}
    const int l    = lane & 15;
    const int kk   = half * 2;

    const float* __restrict__ sArow = &sX[l * STRIDE + kk];
    const float* __restrict__ Wcol  = W + tileN * 16 + l + (size_t)kk * N;

    v8f acc = {};
#pragma unroll
    for (int k0 = 0; k0 < K; k0 += 4) {
        v2f a, b;
        a.x = sArow[k0];
        a.y = sArow[k0 + 1];
        b.x = Wcol[(size_t)k0 * N];
        b.y = Wcol[(size_t)(k0 + 1) * N];
        acc = __builtin_amdgcn_wmma_f32_16x16x4_f32(false, a, false, b, (short)0, acc,
                                                    false, false);
    }

    const float bv      = bias ? bias[tileN * 16 + l] : 0.0f;
    const int   rowBase = tileM * 16 + half * 8;
    float* __restrict__ Zp = Z + (size_t)rowBase * N + tileN * 16 + l;
#pragma unroll
    for (int r = 0; r < 8; ++r) {
        Zp[(size_t)r * N] = acc[r] + bv;
    }
}

// -------------------- edge scatter-add: agg[dst] += T[src] --------------------
// Each thread handles 4 contiguous features of one edge (float4 gather + 4 f32 atomics).
__global__ void aggregate_kernel(const float* __restrict__ T,
                                 const int* __restrict__ src,
                                 const int* __restrict__ dst,
                                 float* __restrict__ agg,
                                 int E, int d) {
    int tid     = blockIdx.x * blockDim.x + threadIdx.x;
    int perEdge = d >> 2;             // threads per edge
    int e       = tid / perEdge;
    if (e >= E) return;
    int c = (tid - e * perEdge) << 2; // feature offset

    int s = src[e];
    int t = dst[e];
    const float4 v = *(const float4*)(T + (size_t)s * d + c);
    float* out = agg + (size_t)t * d + c;
    atomicAdd(out + 0, v.x);
    atomicAdd(out + 1, v.y);
    atomicAdd(out + 2, v.z);
    atomicAdd(out + 3, v.w);
}

// -------------------- combine: out = [relu](S + agg*rdeg), fan-out to 1-2 slots ----------
__global__ void combine_kernel(const float* __restrict__ S,
                               const float* __restrict__ agg,
                               const float* __restrict__ rdeg,
                               float* __restrict__ out1,
                               float* __restrict__ out2,
                               int n, int d, int doRelu) {
    int i = blockIdx.x * blockDim.x + threadIdx.x;
    if (i >= n * d) return;
    int node = i / d;
    float v = fmaf(agg[i], rdeg[node], S[i]);
    if (doRelu) v = fmaxf(v, 0.0f);
    out1[i] = v;
    if (out2) out2[i] = v;
}

extern "C" void kernel_launch(void* const* d_in, const int* in_sizes, int n_in,
                              void* d_out, int out_size, void* d_ws, size_t ws_size,
                              hipStream_t stream) {
    const float* X   = (const float*)d_in[0];
    const int*   src = (const int*)d_in[1];
    const int*   dst = (const int*)d_in[2];
    const float* Ws0 = (const float*)d_in[3];
    const float* Wn0 = (const float*)d_in[4];
    const float* b0  = (const float*)d_in[5];
    const float* Ws1 = (const float*)d_in[6];
    const float* Wn1 = (const float*)d_in[7];
    const float* b1  = (const float*)d_in[8];
    const float* Ws2 = (const float*)d_in[9];
    const float* Wn2 = (const float*)d_in[10];
    const float* b2  = (const float*)d_in[11];
    const int E = in_sizes[1];

    float* out = (float*)d_out;
    // flat output layout in return order: h2 | m0 | m1 | m2 | pen
    const size_t off_m0  = (size_t)N_NODES * D_OUTF;
    const size_t off_m1  = off_m0 + (size_t)N_NODES * D_H;
    const size_t off_m2  = off_m1 + (size_t)N_NODES * D_H;
    const size_t off_pen = off_m2 + (size_t)N_NODES * D_OUTF;

    // workspace: T (neigh-transformed feats) | agg | rdeg
    float* T    = (float*)d_ws;
    float* agg  = T + (size_t)N_NODES * D_H;
    float* rdeg = agg + (size_t)N_NODES * D_H;

    // reciprocal in-degree (shared across layers)
    zero_kernel<<<(N_NODES + 255) / 256, 256, 0, stream>>>(rdeg, N_NODES);
    degree_kernel<<<(E + 255) / 256, 256, 0, stream>>>(dst, rdeg, E);
    recip_kernel<<<(N_NODES + 255) / 256, 256, 0, stream>>>(rdeg, N_NODES);

    constexpr int MTILES = N_NODES / 16;  // 3125, exact

    auto layer128 = [&](const float* x, const float* Wself, const float* Wneigh,
                        const float* bias, float* o1, float* o2, int relu) {
        gemm_wmma_f32<128, 128><<<MTILES, 256, 0, stream>>>(x, Wself, bias, o1);
        gemm_wmma_f32<128, 128><<<MTILES, 256, 0, stream>>>(x, Wneigh, nullptr, T);
        const int nAgg = N_NODES * D_H;
        zero_kernel<<<(nAgg + 255) / 256, 256, 0, stream>>>(agg, nAgg);
        const int aggThreads = E * (D_H / 4);
        aggregate_kernel<<<(aggThreads + 255) / 256, 256, 0, stream>>>(T, src, dst, agg, E, D_H);
        combine_kernel<<<(nAgg + 255) / 256, 256, 0, stream>>>(o1, agg, rdeg, o1, o2,
                                                               N_NODES, D_H, relu);
    };

    // layer 0: h0 -> middle[0]
    layer128(X, Ws0, Wn0, b0, out + off_m0, nullptr, 1);
    // layer 1: h1 -> middle[1] and penultimate
    layer128(out + off_m0, Ws1, Wn1, b1, out + off_m1, out + off_pen, 1);

    // layer 2 (d_out=64, no relu): h2 -> final output and middle[2]
    {
        const float* x = out + off_m1;
        gemm_wmma_f32<128, 64><<<MTILES, 128, 0, stream>>>(x, Ws2, b2, out);
        gemm_wmma_f32<128, 64><<<MTILES, 128, 0, stream>>>(x, Wn2, nullptr, T);
        const int nAgg = N_NODES * D_OUTF;
        zero_kernel<<<(nAgg + 255) / 256, 256, 0, stream>>>(agg, nAgg);
        const int aggThreads = E * (D_OUTF / 4);
        aggregate_kernel<<<(aggThreads + 255) / 256, 256, 0, stream>>>(T, src, dst, agg, E,
                                                                       D_OUTF);
        combine_kernel<<<(nAgg + 255) / 256, 256, 0, stream>>>(out, agg, rdeg, out,
                                                               out + off_m2,
                                                               N_NODES, D_OUTF, 0);
    }
}